// GAT_40295383171129
// MI455X (gfx1250) — compile-verified
//
#include <hip/hip_runtime.h>
#include <math.h>

// ---------------- problem constants (fixed by the reference) ----------------
#define NNODES   100000
#define NEDGES   1600000
#define L1_H     4
#define L1_C     32
#define L1_HC    128        // 4 heads * 32 = layer-1 feature width (also IN_CH)
#define L2_H     1
#define L2_C     64
#define L2_HC    64
#define NEG_SLOPE 0.2f
#define EPS_      1e-16f

typedef __attribute__((ext_vector_type(2))) float v2f;
typedef __attribute__((ext_vector_type(8))) float v8f;

// ---------------- helpers ----------------
__device__ __forceinline__ float atomic_max_f32(float* addr, float val) {
  // Standard monotonic-bits trick; buffer is initialized to -inf.
  if (val >= 0.0f)
    return __int_as_float(atomicMax((int*)addr, __float_as_int(val)));
  return __uint_as_float(atomicMin((unsigned int*)addr, __float_as_uint(val)));
}

__device__ __forceinline__ void edge_nodes(const int* __restrict__ ei, int idx,
                                           int& s, int& d) {
  if (idx < NEDGES) { s = ei[idx]; d = ei[NEDGES + idx]; }   // [2,E] row-major
  else             { s = idx - NEDGES; d = s; }               // implicit self-loop
}

__global__ void fill_kernel(float* __restrict__ p, float v, long long n) {
  long long i = (long long)blockIdx.x * blockDim.x + threadIdx.x;
  if (i < n) p[i] = v;
}

// ---------------- dense GEMM via V_WMMA_F32_16X16X4_F32 ----------------
// C[N x NC] = A[N x K] * B[K x NC], row-major, fp32 exact.
// One wave computes a 16x32 tile (two 16x16 accumulators sharing one A
// fragment -> 3 loads per WMMA instead of 4). Wave w owns cols [32w, 32w+32).
// K, NC are compile-time so the 32-step K-loop fully unrolls (no remainder
// loop, immediate address offsets).
// A layout (ISA 7.12.2, 32-bit A 16x4): lane l holds row M = l&15; VGPR j
// holds K = 2*(l>>4)+j.  B/C/D: one row striped across lanes within a VGPR.
template <int K, int NC>
__global__ __launch_bounds__(128)
void gemm_wmma_kernel(const float* __restrict__ A, const float* __restrict__ B,
                      float* __restrict__ C) {
  const int lane = threadIdx.x & 31;
  const int wv   = threadIdx.x >> 5;
  const int row0 = blockIdx.x * 16;
  const int col0 = wv * 32;
  const int m    = lane & 15;
  const int kk   = (lane >> 4) << 1;          // 0 or 2

  v8f acc0 = {};
  v8f acc1 = {};
  const float* ap = A + (long long)(row0 + m) * K + kk;
  const float* bp = B + (long long)kk * NC + (col0 + m);
#pragma unroll
  for (int k0 = 0; k0 < K; k0 += 4) {
    v2f a, b0, b1;
    a.x  = ap[k0];            a.y  = ap[k0 + 1];            // A rows, K=kk,kk+1
    b0.x = bp[k0 * NC];       b0.y = bp[(k0 + 1) * NC];     // B cols col0+m
    b1.x = bp[k0 * NC + 16];  b1.y = bp[(k0 + 1) * NC + 16];// B cols col0+16+m
    acc0 = __builtin_amdgcn_wmma_f32_16x16x4_f32(
        false, a, false, b0, (short)0, acc0, false, false);
    acc1 = __builtin_amdgcn_wmma_f32_16x16x4_f32(
        false, a, false, b1, (short)0, acc1, false, false);
  }
  // D layout: VGPR r -> M = r + 8*(lane>>4), N = lane&15
  const int orow = row0 + ((lane >> 4) << 3);
  const int ocol = col0 + m;
  float* cp = C + (long long)orow * NC + ocol;
#pragma unroll
  for (int r = 0; r < 8; ++r) {
    cp[(long long)r * NC]      = acc0[r];
    cp[(long long)r * NC + 16] = acc1[r];
  }
}

// ---------------- per-(node,head) attention coefficients ----------------
__global__ void attn_kernel(const float* __restrict__ h,
                            const float* __restrict__ att_s,
                            const float* __restrict__ att_d,
                            float* __restrict__ as_, float* __restrict__ ad_,
                            int H, int C) {
  int i = blockIdx.x * blockDim.x + threadIdx.x;     // over N*H
  if (i >= NNODES * H) return;
  int n = i / H, hd = i % H;
  const float* hv = h + (long long)n * H * C + hd * C;
  const float* s  = att_s + hd * C;
  const float* d  = att_d + hd * C;
  float ss = 0.0f, dd = 0.0f;
  for (int c = 0; c < C; ++c) { float x = hv[c]; ss += x * s[c]; dd += x * d[c]; }
  as_[i] = ss;
  ad_[i] = dd;
}

// ---------------- edge passes (segment softmax + scatter) ----------------
__global__ void edge_max_kernel(const int* __restrict__ ei,
                                const float* __restrict__ as_,
                                const float* __restrict__ ad_,
                                float* __restrict__ mx, int H) {
  int idx = blockIdx.x * blockDim.x + threadIdx.x;
  if (idx >= NEDGES + NNODES) return;
  int s, d; edge_nodes(ei, idx, s, d);
  for (int h = 0; h < H; ++h) {
    float e = as_[s * H + h] + ad_[d * H + h];
    e = e > 0.0f ? e : NEG_SLOPE * e;
    atomic_max_f32(&mx[d * H + h], e);
  }
}

__global__ void edge_sum_kernel(const int* __restrict__ ei,
                                const float* __restrict__ as_,
                                const float* __restrict__ ad_,
                                const float* __restrict__ mx,
                                float* __restrict__ sm, int H) {
  int idx = blockIdx.x * blockDim.x + threadIdx.x;
  if (idx >= NEDGES + NNODES) return;
  int s, d; edge_nodes(ei, idx, s, d);
  for (int h = 0; h < H; ++h) {
    float e = as_[s * H + h] + ad_[d * H + h];
    e = e > 0.0f ? e : NEG_SLOPE * e;
    atomicAdd(&sm[d * H + h], expf(e - mx[d * H + h]));
  }
}

__global__ void edge_scatter_kernel(const int* __restrict__ ei,
                                    const float* __restrict__ as_,
                                    const float* __restrict__ ad_,
                                    const float* __restrict__ mx,
                                    const float* __restrict__ sm,
                                    const float* __restrict__ hfeat,
                                    float* __restrict__ out, int H, int C) {
  const int HC = H * C;
  long long gid = (long long)blockIdx.x * blockDim.x + threadIdx.x;
  long long tot = (long long)(NEDGES + NNODES) * HC;
  if (gid >= tot) return;
  int idx = (int)(gid / HC);
  int c   = (int)(gid % HC);
  int h   = c / C;            // wave-uniform within each 32-channel group
  int s, d; edge_nodes(ei, idx, s, d);
  float e  = as_[s * H + h] + ad_[d * H + h];
  e        = e > 0.0f ? e : NEG_SLOPE * e;
  float ex = expf(e - mx[d * H + h]);
  float alpha = ex / (sm[d * H + h] + EPS_);
  atomicAdd(&out[(long long)d * HC + c], hfeat[(long long)s * HC + c] * alpha);
}

// ---------------- epilogues ----------------
__global__ void bias_elu_kernel(float* __restrict__ o, const float* __restrict__ b,
                                long long n, int HC) {
  long long i = (long long)blockIdx.x * blockDim.x + threadIdx.x;
  if (i >= n) return;
  float v = o[i] + b[(int)(i % HC)];
  o[i] = v > 0.0f ? v : expm1f(v);     // ELU(alpha=1)
}

__global__ void bias_add_kernel(float* __restrict__ o, const float* __restrict__ b,
                                long long n, int HC) {
  long long i = (long long)blockIdx.x * blockDim.x + threadIdx.x;
  if (i >= n) return;
  o[i] += b[(int)(i % HC)];
}

// ---------------- launcher ----------------
extern "C" void kernel_launch(void* const* d_in, const int* in_sizes, int n_in,
                              void* d_out, int out_size, void* d_ws, size_t ws_size,
                              hipStream_t stream) {
  (void)in_sizes; (void)n_in; (void)out_size; (void)ws_size;

  const float* x    = (const float*)d_in[0];
  const int*   ei   = (const int*)  d_in[1];   // [2,E] int32 (JAX x64 disabled)
  const float* W1   = (const float*)d_in[2];
  const float* aS1  = (const float*)d_in[3];
  const float* aD1  = (const float*)d_in[4];
  const float* b1   = (const float*)d_in[5];
  const float* W2   = (const float*)d_in[6];
  const float* aS2  = (const float*)d_in[7];
  const float* aD2  = (const float*)d_in[8];
  const float* b2   = (const float*)d_in[9];
  float* out = (float*)d_out;

  // workspace layout (floats)
  float* ws  = (float*)d_ws;
  float* h1  = ws;                               // N*128 (reused as h2)
  float* o1  = h1  + (long long)NNODES * L1_HC;  // N*128
  float* as1 = o1  + (long long)NNODES * L1_HC;  // N*4
  float* ad1 = as1 + NNODES * L1_H;
  float* mx1 = ad1 + NNODES * L1_H;
  float* sm1 = mx1 + NNODES * L1_H;
  float* as2 = sm1 + NNODES * L1_H;              // N
  float* ad2 = as2 + NNODES;
  float* mx2 = ad2 + NNODES;
  float* sm2 = mx2 + NNODES;
  float* h2  = h1;                               // alias: h1 dead after L1 scatter

  const float NEGINF = -__builtin_inff();
  const int EN = NEDGES + NNODES;
  auto cdiv = [](long long a, long long b) { return (int)((a + b - 1) / b); };

  // init accumulators (deterministic every call)
  fill_kernel<<<cdiv((long long)NNODES * L1_HC, 256), 256, 0, stream>>>(o1, 0.0f, (long long)NNODES * L1_HC);
  fill_kernel<<<cdiv(NNODES * L1_H, 256), 256, 0, stream>>>(mx1, NEGINF, NNODES * L1_H);
  fill_kernel<<<cdiv(NNODES * L1_H, 256), 256, 0, stream>>>(sm1, 0.0f, NNODES * L1_H);
  fill_kernel<<<cdiv(NNODES, 256), 256, 0, stream>>>(mx2, NEGINF, NNODES);
  fill_kernel<<<cdiv(NNODES, 256), 256, 0, stream>>>(sm2, 0.0f, NNODES);
  fill_kernel<<<cdiv((long long)NNODES * L2_HC, 256), 256, 0, stream>>>(out, 0.0f, (long long)NNODES * L2_HC);

  // -------- layer 1: GATConv(128 -> 4 heads x 32, concat) --------
  // 4 waves/block, each owns a 16x32 tile -> 128 cols
  gemm_wmma_kernel<128, L1_HC><<<NNODES / 16, 128, 0, stream>>>(x, W1, h1);
  attn_kernel<<<cdiv(NNODES * L1_H, 128), 128, 0, stream>>>(h1, aS1, aD1, as1, ad1, L1_H, L1_C);
  edge_max_kernel<<<cdiv(EN, 256), 256, 0, stream>>>(ei, as1, ad1, mx1, L1_H);
  edge_sum_kernel<<<cdiv(EN, 256), 256, 0, stream>>>(ei, as1, ad1, mx1, sm1, L1_H);
  edge_scatter_kernel<<<cdiv((long long)EN * L1_HC, 256), 256, 0, stream>>>(
      ei, as1, ad1, mx1, sm1, h1, o1, L1_H, L1_C);
  bias_elu_kernel<<<cdiv((long long)NNODES * L1_HC, 256), 256, 0, stream>>>(
      o1, b1, (long long)NNODES * L1_HC, L1_HC);

  // -------- layer 2: GATConv(128 -> 64, 1 head) --------
  // 2 waves/block, each owns a 16x32 tile -> 64 cols
  gemm_wmma_kernel<128, L2_HC><<<NNODES / 16, 64, 0, stream>>>(o1, W2, h2);
  attn_kernel<<<cdiv(NNODES, 128), 128, 0, stream>>>(h2, aS2, aD2, as2, ad2, L2_H, L2_C);
  edge_max_kernel<<<cdiv(EN, 256), 256, 0, stream>>>(ei, as2, ad2, mx2, L2_H);
  edge_sum_kernel<<<cdiv(EN, 256), 256, 0, stream>>>(ei, as2, ad2, mx2, sm2, L2_H);
  edge_scatter_kernel<<<cdiv((long long)EN * L2_HC, 256), 256, 0, stream>>>(
      ei, as2, ad2, mx2, sm2, h2, out, L2_H, L2_C);
  bias_add_kernel<<<cdiv((long long)NNODES * L2_HC, 256), 256, 0, stream>>>(
      out, b2, (long long)NNODES * L2_HC, L2_HC);
}